// myGCN_48009144435169
// MI455X (gfx1250) — compile-verified
//
#include <hip/hip_runtime.h>
#include <cmath>

typedef __attribute__((ext_vector_type(2))) float v2f;
typedef __attribute__((ext_vector_type(8))) float v8f;

#define IN_DIM 128
#define HID_DIM 128
#define OUT_DIM 64
#define KDIM 128   // inner dim is 128 for both layers

// ---------------------------------------------------------------------------
// Degree / normalization kernels
// ---------------------------------------------------------------------------
__global__ void gcn_fill_one(float* __restrict__ deg, int n) {
    int i = blockIdx.x * blockDim.x + threadIdx.x;
    if (i < n) deg[i] = 1.0f;            // self-loop contributes 1
}

__global__ void gcn_deg_count(const int* __restrict__ dst, float* __restrict__ deg, int e) {
    int i = blockIdx.x * blockDim.x + threadIdx.x;
    if (i < e) unsafeAtomicAdd(&deg[dst[i]], 1.0f);
}

__global__ void gcn_rsqrt_ip(float* __restrict__ deg, int n) {
    int i = blockIdx.x * blockDim.x + threadIdx.x;
    if (i < n) deg[i] = rsqrtf(deg[i]);  // deg >= 1 always (self loops)
}

// ---------------------------------------------------------------------------
// WMMA f32 GEMM: C[M x NOUT] = A[M x 128] * B[128 x NOUT]
// One block per 16-row tile of A (staged via LDS), one wave per 16-col tile.
// Uses V_WMMA_F32_16X16X4_F32 (wave32), layouts per ISA 7.12.2 (32-bit).
// ---------------------------------------------------------------------------
template <int NOUT>
__global__ __launch_bounds__(NOUT * 2) void gcn_gemm_wmma(
    const float* __restrict__ A, const float* __restrict__ B,
    float* __restrict__ C, int M) {
    constexpr int NTHREADS = NOUT * 2;   // (NOUT/16) waves * 32 lanes
    __shared__ float Alds[16 * KDIM];

    const int mTile = blockIdx.x;
    const int tid   = threadIdx.x;

    // Cooperative, coalesced load of the 16x128 A tile (row-clamped for tail).
    for (int i = tid; i < 16 * KDIM; i += NTHREADS) {
        int r = i >> 7;            // i / 128
        int c = i & (KDIM - 1);    // i % 128
        int row = mTile * 16 + r;
        if (row >= M) row = M - 1;
        Alds[i] = A[(size_t)row * KDIM + c];
    }
    __syncthreads();

    const int wave  = tid >> 5;
    const int lane  = tid & 31;
    const int lrow  = lane & 15;   // A-matrix M index for this lane
    const int khalf = lane >> 4;   // 0: K+0/K+1, 1: K+2/K+3
    const int col   = lrow + wave * 16;

    v8f acc = {};
    #pragma unroll
    for (int k0 = 0; k0 < KDIM; k0 += 4) {
        const int ka = k0 + 2 * khalf;
        v2f a, b;
        a.x = Alds[lrow * KDIM + ka];
        a.y = Alds[lrow * KDIM + ka + 1];
        b.x = B[(size_t)ka * NOUT + col];
        b.y = B[(size_t)(ka + 1) * NOUT + col];
        acc = __builtin_amdgcn_wmma_f32_16x16x4_f32(
            /*neg_a=*/false, a, /*neg_b=*/false, b,
            /*c_mod=*/(short)0, acc, /*reuse_a=*/false, /*reuse_b=*/false);
    }

    // C layout: lanes 0-15 -> rows 0..7 (vgpr idx), lanes 16-31 -> rows 8..15.
    const int rowBase = mTile * 16 + 8 * khalf;
    #pragma unroll
    for (int i = 0; i < 8; ++i) {
        int row = rowBase + i;
        if (row < M) C[(size_t)row * NOUT + col] = acc[i];
    }
}

// ---------------------------------------------------------------------------
// agg[n][f] = h[n][f] * dinv[n]^2   (self-loop term; also fully inits buffer)
// ---------------------------------------------------------------------------
template <int F>
__global__ void gcn_selfloop_init(const float* __restrict__ H,
                                  const float* __restrict__ dinv,
                                  float* __restrict__ agg, size_t total) {
    size_t i = (size_t)blockIdx.x * blockDim.x + threadIdx.x;
    if (i >= total) return;
    int n = (int)(i / F);
    float di = dinv[n];
    agg[i] = H[i] * di * di;
}

// ---------------------------------------------------------------------------
// Per-edge scatter: one wave per edge, F/32 features per lane.
// H rows (51MB) and agg (<=51MB) are L2-resident on MI455X (192MB L2),
// so gathers + fp32 atomics stay on-chip.
// ---------------------------------------------------------------------------
template <int F>
__global__ void gcn_edge_scatter(const float* __restrict__ H,
                                 const int* __restrict__ src,
                                 const int* __restrict__ dst,
                                 const float* __restrict__ dinv,
                                 float* __restrict__ agg, int E) {
    const int lane = threadIdx.x & 31;
    const int e = blockIdx.x * (blockDim.x >> 5) + (threadIdx.x >> 5);
    if (e >= E) return;
    const int s = src[e];
    const int d = dst[e];
    const float w = dinv[s] * dinv[d];
    const float* hp = H + (size_t)s * F;
    float* ap = agg + (size_t)d * F;
    #pragma unroll
    for (int j = 0; j < F / 32; ++j) {
        int f = lane + j * 32;
        unsafeAtomicAdd(ap + f, hp[f] * w);
    }
}

// ---------------------------------------------------------------------------
// out[i] = elu(agg[i] + bias[i % F])   (in-place safe)
// ---------------------------------------------------------------------------
template <int F>
__global__ void gcn_bias_elu(const float* __restrict__ agg,
                             const float* __restrict__ bias,
                             float* __restrict__ out, size_t total) {
    size_t i = (size_t)blockIdx.x * blockDim.x + threadIdx.x;
    if (i >= total) return;
    float v = agg[i] + bias[i & (F - 1)];
    out[i] = (v > 0.0f) ? v : expm1f(v);
}

// ---------------------------------------------------------------------------
// Host-side launch
// ---------------------------------------------------------------------------
extern "C" void kernel_launch(void* const* d_in, const int* in_sizes, int n_in,
                              void* d_out, int out_size, void* d_ws, size_t ws_size,
                              hipStream_t stream) {
    const float* x  = (const float*)d_in[0];
    const int*   ei = (const int*)d_in[1];
    const float* W1 = (const float*)d_in[2];
    const float* b1 = (const float*)d_in[3];
    const float* W2 = (const float*)d_in[4];
    const float* b2 = (const float*)d_in[5];
    float* out = (float*)d_out;

    const int N = in_sizes[0] / IN_DIM;   // 100000
    const int E = in_sizes[1] / 2;        // 1600000
    const int* src = ei;
    const int* dst = ei + E;

    // Workspace layout
    char* ws = (char*)d_ws;
    float* dinv = (float*)ws;                                       // N floats
    size_t off1 = ((size_t)N * 4 + 255) & ~(size_t)255;
    float* hbuf = (float*)(ws + off1);                              // N*128 floats (h1, later h2)
    size_t off2 = off1 + ((((size_t)N * HID_DIM * 4) + 255) & ~(size_t)255);
    float* agg1 = (float*)(ws + off2);                              // N*128 floats (agg1 -> act1)

    const int TB = 256;
    const int edgeBlocks = (E + 7) / 8;          // 8 edges (waves) per 256-thread block
    const int mTiles = (N + 15) / 16;
    const size_t tot1 = (size_t)N * HID_DIM;
    const size_t tot2 = (size_t)N * OUT_DIM;

    // ---- degrees / symmetric normalization ----
    gcn_fill_one<<<(N + TB - 1) / TB, TB, 0, stream>>>(dinv, N);
    gcn_deg_count<<<(E + TB - 1) / TB, TB, 0, stream>>>(dst, dinv, E);
    gcn_rsqrt_ip<<<(N + TB - 1) / TB, TB, 0, stream>>>(dinv, N);

    // ---- layer 1: h1 = x @ W1 ; aggregate ; elu(agg + b1) in place ----
    gcn_gemm_wmma<HID_DIM><<<mTiles, HID_DIM * 2, 0, stream>>>(x, W1, hbuf, N);
    gcn_selfloop_init<HID_DIM><<<(int)((tot1 + TB - 1) / TB), TB, 0, stream>>>(hbuf, dinv, agg1, tot1);
    gcn_edge_scatter<HID_DIM><<<edgeBlocks, TB, 0, stream>>>(hbuf, src, dst, dinv, agg1, E);
    gcn_bias_elu<HID_DIM><<<(int)((tot1 + TB - 1) / TB), TB, 0, stream>>>(agg1, b1, agg1, tot1);

    // ---- layer 2: h2 = act1 @ W2 (into hbuf); agg2 accumulates in d_out ----
    gcn_gemm_wmma<OUT_DIM><<<mTiles, OUT_DIM * 2, 0, stream>>>(agg1, W2, hbuf, N);
    gcn_selfloop_init<OUT_DIM><<<(int)((tot2 + TB - 1) / TB), TB, 0, stream>>>(hbuf, dinv, out, tot2);
    gcn_edge_scatter<OUT_DIM><<<edgeBlocks, TB, 0, stream>>>(hbuf, src, dst, dinv, out, E);
    gcn_bias_elu<OUT_DIM><<<(int)((tot2 + TB - 1) / TB), TB, 0, stream>>>(out, b2, out, tot2);
}